// SSIM_3135326126170
// MI455X (gfx1250) — compile-verified
//
#include <hip/hip_runtime.h>
#include <hip/hip_bf16.h>

// ---------------- problem constants ----------------
#define DIN   192
#define DOUT  182
#define WSZ   11

static constexpr int    PLANE = DIN * DIN;               // 36864
static constexpr int    FSZ   = DOUT * PLANE;            // 6,709,248  bufZ field: (z',y,x)
static constexpr int    FSY   = DOUT * DOUT * DIN;       // 6,359,808  bufY field: (z',y',x)
static constexpr long long NVOX = 2LL * DOUT * DOUT * DOUT; // 12,057,136 outputs in mean

// Gaussian(sigma=1.5, 11 taps), normalized (matches reference float64->float32)
__constant__ float c_w[WSZ] = {
    0.00102838f, 0.00759876f, 0.03600076f, 0.10936069f, 0.21300538f,
    0.26601260f,
    0.21300538f, 0.10936069f, 0.03600076f, 0.00759876f, 0.00102838f};

typedef __attribute__((ext_vector_type(2))) float v2f;
typedef __attribute__((ext_vector_type(8))) float v8f;
typedef int v4i_vs __attribute__((vector_size(16)));   // matches builtin param pointee

#if defined(__gfx1250__) && __has_builtin(__builtin_amdgcn_wmma_f32_16x16x4_f32)
#define HAVE_WMMA_F32X4 1
#endif

// ---- CDNA5 async global->LDS copy (ASYNCcnt-tracked, no VGPR staging) ----
__device__ __forceinline__ void async_copy_b128(const float* g, float* l) {
#if defined(__gfx1250__)
#if __has_builtin(__builtin_amdgcn_global_load_async_to_lds_b128)
    typedef __attribute__((address_space(1))) v4i_vs* gp_t;
    typedef __attribute__((address_space(3))) v4i_vs* lp_t;
    __builtin_amdgcn_global_load_async_to_lds_b128(
        (gp_t)g, (lp_t)l, /*imm offset=*/0, /*cpol=*/0);
#else
    unsigned lds_off =
        (unsigned)(size_t)(__attribute__((address_space(3))) float*)l;
    asm volatile("global_load_async_to_lds_b128 %0, %1, off"
                 :: "v"(lds_off), "v"(g) : "memory");
#endif
#else
    // host-side parse / non-gfx1250: synchronous copy
    for (int i = 0; i < 4; ++i) l[i] = g[i];
#endif
}

__device__ __forceinline__ void wait_async0() {
#if defined(__gfx1250__)
#if __has_builtin(__builtin_amdgcn_s_wait_asynccnt)
    __builtin_amdgcn_s_wait_asynccnt(0);
#else
    asm volatile("s_wait_asynccnt 0" ::: "memory");
#endif
#endif
}

// safe tap lookup: 0 outside [0,10]; load is always in-bounds, then select
__device__ __forceinline__ float wtap(int i) {
    int j = min(max(i, 0), WSZ - 1);
    float w = c_w[j];
    return (i == j) ? w : 0.0f;
}

// order-preserving float<->uint key (total order incl. negatives)
__device__ __forceinline__ unsigned fkey(float f) {
    unsigned u = __float_as_uint(f);
    return (u & 0x80000000u) ? ~u : (u | 0x80000000u);
}
__device__ __forceinline__ float funkey(unsigned k) {
    unsigned u = (k & 0x80000000u) ? (k & 0x7FFFFFFFu) : ~k;
    return __uint_as_float(u);
}

// ---------------- init ----------------
__global__ void k_init(double* acc, unsigned* keys) {
    acc[0]  = 0.0;
    keys[0] = 0u;           // running max key
    keys[1] = 0xFFFFFFFFu;  // running min key
}

// ---------------- min/max of img1 (for dynamic range L) ----------------
__global__ void k_minmax(const float* __restrict__ x, long long n, unsigned* keys) {
    __shared__ unsigned smax[256], smin[256];
    unsigned kmax = 0u, kmin = 0xFFFFFFFFu;
    for (long long i = (long long)blockIdx.x * 256 + threadIdx.x; i < n;
         i += 256LL * gridDim.x) {
        unsigned k = fkey(x[i]);
        kmax = max(kmax, k);
        kmin = min(kmin, k);
    }
    smax[threadIdx.x] = kmax; smin[threadIdx.x] = kmin;
    __syncthreads();
    for (int s = 128; s > 0; s >>= 1) {
        if (threadIdx.x < (unsigned)s) {
            smax[threadIdx.x] = max(smax[threadIdx.x], smax[threadIdx.x + s]);
            smin[threadIdx.x] = min(smin[threadIdx.x], smin[threadIdx.x + s]);
        }
        __syncthreads();
    }
    if (threadIdx.x == 0) {
        atomicMax(&keys[0], smax[0]);
        atomicMin(&keys[1], smin[0]);
    }
}

// ---------------- pass 1: z-axis 11-tap conv of {a, b, a^2, b^2, ab} ----------------
__global__ void k_pass1_z(const float* __restrict__ img1,
                          const float* __restrict__ img2,
                          float* __restrict__ bufZ) {
    size_t idx = (size_t)blockIdx.x * 256 + threadIdx.x;
    if (idx >= (size_t)FSZ) return;
    // streaming hint (global_prefetch_b8)
    __builtin_prefetch(img1 + idx + 10 * (size_t)PLANE, 0, 1);
    __builtin_prefetch(img2 + idx + 10 * (size_t)PLANE, 0, 1);
    float s0 = 0.f, s1 = 0.f, s2 = 0.f, s3 = 0.f, s4 = 0.f;
#pragma unroll
    for (int t = 0; t < WSZ; ++t) {
        float a = img1[idx + (size_t)t * PLANE];
        float b = img2[idx + (size_t)t * PLANE];
        float w = c_w[t];
        s0 += w * a;        s1 += w * b;
        s2 += w * (a * a);  s3 += w * (b * b);  s4 += w * (a * b);
    }
    bufZ[idx]                   = s0;
    bufZ[idx + 1 * (size_t)FSZ] = s1;
    bufZ[idx + 2 * (size_t)FSZ] = s2;
    bufZ[idx + 3 * (size_t)FSZ] = s3;
    bufZ[idx + 4 * (size_t)FSZ] = s4;
}

// ---------------- pass 2: y-axis conv as Toeplitz matmul via WMMA f32 16x16x4 ----------
// Out(16 y' x 16 x) tile = sum_c A_c(16x4 Toeplitz taps) * B_c(4x16 input rows)
__global__ void k_pass2_y(const float* __restrict__ bufZ, float* __restrict__ bufY) {
    const int lane = threadIdx.x;          // 0..31, one wave per block
    int tile = blockIdx.x;
    const int xt = tile % 12;  tile /= 12;           // 12 x-tiles (192/16)
    const int yt = tile % 12;  tile /= 12;           // 12 y'-tiles (ceil(182/16))
    const int zp = tile % DOUT; tile /= DOUT;        // 182 z' planes
    const int f  = tile;                             // 5 fields

    const float* __restrict__ src = bufZ + (size_t)f * FSZ + (size_t)zp * PLANE;
    float* __restrict__ dst       = bufY + (size_t)f * FSY + (size_t)zp * (DOUT * DIN);

    const int x0 = xt * 16, y0 = yt * 16;
    const int n  = lane & 15;   // N column (x) / M row (for A)
    const int hi = lane >> 4;   // half-wave selector

#ifdef HAVE_WMMA_F32X4
    v8f acc = {0.f, 0.f, 0.f, 0.f, 0.f, 0.f, 0.f, 0.f};
#pragma unroll
    for (int c = 0; c < 7; ++c) {           // K = 28 >= band width 26
        const int kb = 4 * c + 2 * hi;
        // A: Toeplitz taps, A[m][k] = w[k_global - m], m = lane&15
        v2f A; A.x = wtap(kb - n); A.y = wtap(kb + 1 - n);
        // B: input rows y0+kb, y0+kb+1 at columns x0..x0+15 (clamped, weights mask OOB)
        const int ra = min(y0 + kb,     DIN - 1);
        const int rb = min(y0 + kb + 1, DIN - 1);
        v2f B; B.x = src[ra * DIN + x0 + n]; B.y = src[rb * DIN + x0 + n];
        acc = __builtin_amdgcn_wmma_f32_16x16x4_f32(
            /*neg_a=*/false, A, /*neg_b=*/false, B,
            /*c_mod=*/(short)0, acc, /*reuse_a=*/false, /*reuse_b=*/false);
    }
#pragma unroll
    for (int v = 0; v < 8; ++v) {           // D layout: M = v + 8*hi, N = lane&15
        const int yo = y0 + v + 8 * hi;
        if (yo < DOUT) dst[yo * DIN + x0 + n] = acc[v];
    }
#else
    // scalar fallback (also used by host-side parse): identical math
    float accv[8];
#pragma unroll
    for (int v = 0; v < 8; ++v) {
        const int yo = y0 + v + 8 * hi;
        float s = 0.f;
        if (yo < DOUT) {
#pragma unroll
            for (int t = 0; t < WSZ; ++t) s += c_w[t] * src[(yo + t) * DIN + x0 + n];
        }
        accv[v] = s;
    }
#pragma unroll
    for (int v = 0; v < 8; ++v) {
        const int yo = y0 + v + 8 * hi;
        if (yo < DOUT) dst[yo * DIN + x0 + n] = accv[v];
    }
#endif
}

// ---------------- pass 3: async LDS staging + x-axis conv + SSIM + reduction --------
__global__ void k_pass3_x(const float* __restrict__ bufY, double* acc,
                          const unsigned* __restrict__ keys) {
    __shared__ __align__(16) float lds[5][DIN];   // 5 fields x 192 floats = 3840 B
    __shared__ float red[256];
    const int t = threadIdx.x;                // 0..255
    const int y = blockIdx.x;                 // 0..181 (y')
    const int z = blockIdx.y;                 // 0..181 (z')
    const size_t base = (size_t)z * (DOUT * DIN) + (size_t)y * DIN;

    // 240 lanes x 16B async copies: memory -> LDS without VGPR staging
    if (t < 240) {
        const int f = t / 48;                 // field 0..4
        const int c = t % 48;                 // 16-byte chunk within the 768B row
        async_copy_b128(bufY + (size_t)f * FSY + base + (size_t)c * 4,
                        &lds[f][c * 4]);
    }
    wait_async0();        // each wave drains its own ASYNCcnt
    __syncthreads();      // cross-wave LDS visibility

    float val = 0.f;
    if (t < DOUT) {
        float s0 = 0.f, s1 = 0.f, s2 = 0.f, s3 = 0.f, s4 = 0.f;
#pragma unroll
        for (int k = 0; k < WSZ; ++k) {
            float w = c_w[k];
            s0 += w * lds[0][t + k]; s1 += w * lds[1][t + k];
            s2 += w * lds[2][t + k]; s3 += w * lds[3][t + k];
            s4 += w * lds[4][t + k];
        }
        const float maxv = funkey(keys[0]);
        const float minv = funkey(keys[1]);
        const float L  = (maxv > 128.f ? 255.f : 1.f) - (minv < -0.5f ? -1.f : 0.f);
        const float C1 = (0.01f * L) * (0.01f * L);
        const float C2 = (0.03f * L) * (0.03f * L);
        const float mu1 = s0, mu2 = s1;
        const float mu1s = mu1 * mu1, mu2s = mu2 * mu2, mu12 = mu1 * mu2;
        const float sg1 = s2 - mu1s, sg2 = s3 - mu2s, sg12 = s4 - mu12;
        const float v1 = 2.f * sg12 + C2, v2 = sg1 + sg2 + C2;
        val = (2.f * mu12 + C1) * v1 / ((mu1s + mu2s + C1) * v2);
    }
    red[t] = val;
    __syncthreads();
    for (int s = 128; s > 0; s >>= 1) {
        if (t < s) red[t] += red[t + s];
        __syncthreads();
    }
    if (t == 0) atomicAdd(acc, (double)red[0]);   // global_atomic_add_f64
}

// ---------------- final: mean ----------------
__global__ void k_final(const double* acc, float* out) {
    out[0] = (float)(acc[0] / (double)NVOX);
}

extern "C" void kernel_launch(void* const* d_in, const int* in_sizes, int n_in,
                              void* d_out, int out_size, void* d_ws, size_t ws_size,
                              hipStream_t stream) {
    const float* img1 = (const float*)d_in[0];  // (2,1,192,192,192) f32
    const float* img2 = (const float*)d_in[1];
    float* out = (float*)d_out;

    // ws layout: [double acc][2x uint minmax keys][bufZ 5*FSZ f32][bufY 5*FSY f32]
    // requires ~262 MB of workspace
    double*   acc  = (double*)d_ws;
    unsigned* keys = (unsigned*)d_ws + 2;
    float*    bufZ = (float*)d_ws + 8;
    float*    bufY = bufZ + 5 * (size_t)FSZ;

    k_init<<<1, 1, 0, stream>>>(acc, keys);
    k_minmax<<<1024, 256, 0, stream>>>(img1, 2LL * DIN * DIN * DIN, keys);

    for (int b = 0; b < 2; ++b) {
        const float* a  = img1 + (size_t)b * DIN * DIN * DIN;
        const float* bb = img2 + (size_t)b * DIN * DIN * DIN;
        k_pass1_z<<<(FSZ + 255) / 256, 256, 0, stream>>>(a, bb, bufZ);
        k_pass2_y<<<12 * 12 * DOUT * 5, 32, 0, stream>>>(bufZ, bufY);
        k_pass3_x<<<dim3(DOUT, DOUT), 256, 0, stream>>>(bufY, acc, keys);
    }
    k_final<<<1, 1, 0, stream>>>(acc, out);
}